// DiffTreeInterpreter_58669253263510
// MI455X (gfx1250) — compile-verified
//
#include <hip/hip_runtime.h>

// DiffTreeInterpreter on MI455X (gfx1250).
//
// Scatter-add problem, not a GEMM: ~0.2 GFLOP against ~460 MB of traffic
// (AI ~0.4 FLOP/B); WMMA has no dense tile to multiply (one-hot scatter,
// ~2 items/bin). MI455X-aware plan:
//   * output (64 MB) << L2 (192 MB): all f32 atomics resolve in L2 as
//     non-returning GLOBAL_ATOMIC_ADD_F32 (STOREcnt-tracked)
//   * mem_values (128 MB) + index arrays are stream-once: nontemporal loads
//     (th:TH_LOAD_NT) so they don't evict the atomic working set from L2
//   * wave32-per-item, float4 per lane -> 512B fully-coalesced row loads
// Floor: (128MB NT read + 64MB init + 64MB wb) / 23.3 TB/s ~= 11 us.

namespace {

constexpr int B = 32;
constexpr int L = 128;
constexpr int F = 128;        // 2^7
constexpr int R = 4096;       // R*F = 2^19
constexpr int N = 262144;
constexpr int H = R >> 1;     // 2048

typedef float v4f __attribute__((ext_vector_type(4)));

__device__ __forceinline__ void atomic_add_f32(float* p, float v) {
  // Relaxed device-scope fadd; result discarded so the backend selects the
  // non-returning GLOBAL_ATOMIC_ADD_F32.
  (void)__hip_atomic_fetch_add(p, v, __ATOMIC_RELAXED, __HIP_MEMORY_SCOPE_AGENT);
}

// Defined FIRST so the disasm snippet shows this kernel's atomic lowering.
// One wave32 per item; lane k owns floats [4k, 4k+4) of the 128-float row.
// Per-item scalars/branches are wave-uniform; broadcast loads coalesce.
__global__ __launch_bounds__(256) void scatter_kernel(
    const float* __restrict__ mem,
    const float* __restrict__ arg_w,
    const float* __restrict__ op_dist,
    const int* __restrict__ batch_idx,
    const int* __restrict__ slot_idx,
    const int* __restrict__ role_idx,
    float* __restrict__ out) {
  const int item = (blockIdx.x * 256 + threadIdx.x) >> 5;
  const int lane = threadIdx.x & 31;
  if (item >= N) return;

  // Stream-once index data: nontemporal.
  const int b = __builtin_nontemporal_load(batch_idx + item);
  const int s = __builtin_nontemporal_load(slot_idx + item);
  const int r = __builtin_nontemporal_load(role_idx + item);

  // Reused small tables: regular temporal loads.
  const v4f w =
      *reinterpret_cast<const v4f*>(arg_w + (((b << 7) + s) << 2));
  const float op0 = op_dist[b * 3 + 0];
  const float op1 = op_dist[b * 3 + 1];
  const float op2 = op_dist[b * 3 + 2];

  // Stream-once 128 MB payload: nontemporal b128 load.
  const int fo = lane << 2;
  const v4f v = __builtin_nontemporal_load(
      reinterpret_cast<const v4f*>(mem + (size_t)item * F + fo));

  float* base = out + ((size_t)b << 19);

  // car (r even) / cdr (r odd, r > 1): destination role r>>1.
  if (r != 1) {
    const float sa = (r & 1) ? op1 * w.y : op0 * w.x;
    float* d = base + ((r >> 1) << 7) + fo;
    atomic_add_f32(d + 0, sa * v.x);
    atomic_add_f32(d + 1, sa * v.y);
    atomic_add_f32(d + 2, sa * v.z);
    atomic_add_f32(d + 3, sa * v.w);
  }

  // cons: r < H -> roles 2r (op2*w2) and 2r+1 (op2*w3).
  // The reference's `pad` mask is a numerical no-op (pad rows are all-zero).
  if (r < H) {
    const float s1 = op2 * w.z;
    const float s2 = op2 * w.w;
    float* d1 = base + ((r << 1) << 7) + fo;
    atomic_add_f32(d1 + 0, s1 * v.x);
    atomic_add_f32(d1 + 1, s1 * v.y);
    atomic_add_f32(d1 + 2, s1 * v.z);
    atomic_add_f32(d1 + 3, s1 * v.w);
    float* d2 = d1 + F;
    atomic_add_f32(d2 + 0, s2 * v.x);
    atomic_add_f32(d2 + 1, s2 * v.y);
    atomic_add_f32(d2 + 2, s2 * v.z);
    atomic_add_f32(d2 + 3, s2 * v.w);
  }
}

// out = 0 everywhere, except out[b, 1, :] = op2[b] * root_filler[b, :].
// One float4 per thread; all 4 elements of a float4 share the same (b, j).
// Regular-temporal stores: we WANT the output resident/dirty in L2.
__global__ __launch_bounds__(256) void init_out_kernel(
    const float* __restrict__ root_filler,
    const float* __restrict__ op_dist,
    float* __restrict__ out) {
  const int t = blockIdx.x * 256 + threadIdx.x;
  const int e = t << 2;                 // element offset
  const int b = e >> 19;                // / (R*F)
  const int rem = e & ((R * F) - 1);
  const int j = rem >> 7;               // / F
  const int f = rem & (F - 1);
  v4f v = {0.f, 0.f, 0.f, 0.f};
  if (j == 1) {
    const float op2 = op_dist[b * 3 + 2];
    const v4f rf = *reinterpret_cast<const v4f*>(root_filler + b * F + f);
    v = op2 * rf;
  }
  *reinterpret_cast<v4f*>(out + e) = v;
}

}  // namespace

extern "C" void kernel_launch(void* const* d_in, const int* in_sizes, int n_in,
                              void* d_out, int out_size, void* d_ws,
                              size_t ws_size, hipStream_t stream) {
  const float* mem = (const float*)d_in[0];   // (N, F) f32
  const float* aw  = (const float*)d_in[1];   // (B, L, 4) f32
  const float* rf  = (const float*)d_in[2];   // (B, F) f32
  const float* od  = (const float*)d_in[3];   // (B, 3) f32
  const int* bidx  = (const int*)d_in[4];     // (N,) i32
  const int* sidx  = (const int*)d_in[5];     // (N,) i32
  const int* ridx  = (const int*)d_in[6];     // (N,) i32
  float* out = (float*)d_out;                 // (B, R, F) f32

  constexpr int kInitThreads = (B * R * F) / 4;     // 4,194,304
  init_out_kernel<<<kInitThreads / 256, 256, 0, stream>>>(rf, od, out);

  constexpr int kWavesPerBlock = 256 / 32;          // 8 items per block
  scatter_kernel<<<N / kWavesPerBlock, 256, 0, stream>>>(mem, aw, od, bidx,
                                                         sidx, ridx, out);
}